// PiecewiseDiscontinuousPolynomial_5257039970367
// MI455X (gfx1250) — compile-verified
//
#include <hip/hip_runtime.h>
#include <stdint.h>

// Problem constants (from reference)
#define N_NODES   5
#define SEGMENTS  32
#define NFEAT     256
#define NBATCH    32768
#define WCOLS     (NFEAT * N_NODES * SEGMENTS)   // 40960 (full w row length)
#define USED_COLS (N_NODES * SEGMENTS)           // 160 columns actually used
// TDM pads 1 DWORD every 32 DWORDs -> 160-DWORD row becomes 165-DWORD stride.
// 165 mod 64 = 37 (odd) -> consecutive-f lane reads are bank-conflict-free.
#define ROW_STRIDE_DW 165
#define LDS_DW (NFEAT * ROW_STRIDE_DW)           // 42240 dwords = 168960 B
static_assert(LDS_DW * 4 <= 320 * 1024, "exceeds WGP LDS");

typedef unsigned int v4u __attribute__((ext_vector_type(4)));
typedef int          v8i __attribute__((ext_vector_type(8)));
typedef int          v4i __attribute__((ext_vector_type(4)));

__global__ __launch_bounds__(1024, 1)
void pdp_tdm_kernel(const float* __restrict__ x, const float* __restrict__ w,
                    float* __restrict__ out, int rows_per_block)
{
    __shared__ float wlds[LDS_DW];

    const int tid = threadIdx.x;

    // ---- Stage w[0:256, 0:160] -> LDS via Tensor Data Mover (wave 0 only) ----
    if (tid < 32) {
        const uint64_t gaddr = (uint64_t)(uintptr_t)w;
        const uint32_t laddr = (uint32_t)(uintptr_t)(void*)wlds; // low 32b of flat = LDS byte addr

        v4u g0;
        g0.x = 1u;                                    // count=1 (valid descriptor)
        g0.y = laddr;                                 // lds_addr
        g0.z = (uint32_t)(gaddr & 0xffffffffu);       // global_addr[31:0]
        g0.w = (uint32_t)((gaddr >> 32) & 0x01ffffffu) | (2u << 30); // addr[56:32] | type=2

        v8i g1;
        g1[0] = (int)((2u << 16)       // data_size = 4B
                    | (1u << 20)       // pad_enable
                    | (4u << 22)       // pad_interval code 4 -> every 32 DWORDs
                    | (0u << 25));     // pad_amount  code 0 -> 1 DWORD
        g1[1] = (int)(((uint32_t)(WCOLS & 0xffff)) << 16);     // tensor_dim0[15:0]=40960
        g1[2] = (int)(((uint32_t)WCOLS >> 16)                  // tensor_dim0[31:16]
                    | (((uint32_t)NFEAT & 0xffff) << 16));     // tensor_dim1[15:0]=256
        g1[3] = (int)(((uint32_t)USED_COLS) << 16);            // tile_dim0 = 160
        g1[4] = (int)NFEAT;                                    // tile_dim1 = 256, tile_dim2 = 0
        g1[5] = (int)WCOLS;                                    // tensor_dim0_stride[31:0] = 40960
        g1[6] = 0;                                             // stride hi / dim1_stride lo
        g1[7] = 0;

        v4i gz = {0, 0, 0, 0};
#if __clang_major__ >= 23
        v8i gz8 = {0, 0, 0, 0, 0, 0, 0, 0};
        __builtin_amdgcn_tensor_load_to_lds(g0, g1, gz, gz, gz8, 0);
#else
        __builtin_amdgcn_tensor_load_to_lds(g0, g1, gz, gz, 0);
#endif
        __builtin_amdgcn_s_wait_tensorcnt(0);
    }
    __syncthreads();

    // ---- Streaming evaluation: 1024 threads = 4 rows x 256 features ----
    const int f    = tid & (NFEAT - 1);
    const int rsub = tid >> 8;                       // 0..3
    const int row0 = blockIdx.x * rows_per_block;
    const int wrow = f * ROW_STRIDE_DW;              // this feature's LDS row (dwords)

#pragma unroll 2
    for (int it = 0; it < rows_per_block; it += 4) {
        const int row  = row0 + it + rsub;
        const int urow = __builtin_amdgcn_readfirstlane(row); // wave-uniform row

        // per-sample segment from feature 0 (scalar path)
        const float x0 = x[(size_t)urow * NFEAT];
        int s0 = (int)floorf((x0 + 1.0f) * 0.5f * (float)SEGMENTS);
        s0 = s0 < 0 ? 0 : (s0 > SEGMENTS - 1 ? SEGMENTS - 1 : s0);
        const int wid0 = s0 * N_NODES;

        // local coordinate (bit-exact vs reference: x_max-x_min == 2/SEG exactly)
        const float xv    = x[(size_t)urow * NFEAT + f];
        const float idf   = floorf((xv + 1.0f) * 0.5f * (float)SEGMENTS);
        const float x_min = idf * (2.0f / (float)SEGMENTS) - 1.0f;
        const float xi    = (xv - x_min) * (float)SEGMENTS - 1.0f;

        // Lagrange basis on nodes {-1,-0.5,0,0.5,1}
        const float d0 = xi + 1.0f;
        const float d1 = xi + 0.5f;
        const float d2 = xi;
        const float d3 = xi - 0.5f;
        const float d4 = xi - 1.0f;
        const float q01 = d0 * d1;
        const float q34 = d3 * d4;
        const float b0 = ( 2.0f / 3.0f) * d1 * d2 * q34;
        const float b1 = (-8.0f / 3.0f) * d0 * d2 * q34;
        const float b2 =   4.0f         * q01 * q34;
        const float b3 = (-8.0f / 3.0f) * q01 * d2 * d4;
        const float b4 = ( 2.0f / 3.0f) * q01 * d2 * d3;

        // padded-LDS weight fetch: addr = f*165 + c + (c>>5)
        const int c0 = wid0,     c1 = wid0 + 1, c2 = wid0 + 2,
                  c3 = wid0 + 3, c4 = wid0 + 4;
        const float w0v = wlds[wrow + c0 + (c0 >> 5)];
        const float w1v = wlds[wrow + c1 + (c1 >> 5)];
        const float w2v = wlds[wrow + c2 + (c2 >> 5)];
        const float w3v = wlds[wrow + c3 + (c3 >> 5)];
        const float w4v = wlds[wrow + c4 + (c4 >> 5)];

        const float fx = b0 * w0v + b1 * w1v + b2 * w2v + b3 * w3v + b4 * w4v;
        out[(size_t)row * NFEAT + f] = fx;
    }
}

extern "C" void kernel_launch(void* const* d_in, const int* in_sizes, int n_in,
                              void* d_out, int out_size, void* d_ws, size_t ws_size,
                              hipStream_t stream) {
    (void)in_sizes; (void)n_in; (void)out_size; (void)d_ws; (void)ws_size;
    const float* x = (const float*)d_in[0];   // (32768, 256)
    const float* w = (const float*)d_in[1];   // (256, 40960)
    float* out = (float*)d_out;               // (32768, 256)

    const int blocks = 512;                   // 1 WG/WGP (LDS-limited); cover large WGP counts
    const int rows_per_block = NBATCH / blocks; // 64, multiple of 4
    pdp_tdm_kernel<<<dim3(blocks), dim3(1024), 0, stream>>>(x, w, out, rows_per_block);
}